// PointGraphic2d_44684839748320
// MI455X (gfx1250) — compile-verified
//
#include <hip/hip_runtime.h>

typedef __attribute__((ext_vector_type(2))) float v2f;
typedef __attribute__((ext_vector_type(4))) float f4;
typedef __attribute__((ext_vector_type(8))) float v8f;

#define CANVAS_H 4096
#define CANVAS_W 4096
#define LINE_WIDTH 20.0f
#define EPSILON_ 0.001f

__global__ __launch_bounds__(256) void PointGraphic2d_kernel(
    const float* __restrict__ kp, float* __restrict__ out) {
  const int lane = threadIdx.x & 31;
  const int wave = threadIdx.x >> 5;
  const int tile_x = blockIdx.x * 8 + wave;  // 8 waves per block, 16 px per tile
  const int tile_y = blockIdx.y;
  const int x0 = tile_x * 16;
  const int y0 = tile_y * 16;

  // Point in pixel coords (uniform; tiny cached load)
  const float py = kp[0] * (float)CANVAS_H;
  const float px = kp[1] * (float)CANVAS_W;

  // Min distance^2 from the point to this tile's bounding box (wave-uniform).
  float ddx = fmaxf(fmaxf((float)x0 - px, px - (float)(x0 + 15)), 0.0f);
  float ddy = fmaxf(fmaxf((float)y0 - py, py - (float)(y0 + 15)), 0.0f);
  float box_d2 = ddx * ddx + ddy * ddy;

  // Uniform branch: keep EXEC all-ones (required for WMMA) and let the
  // compiler use s_cbranch instead of exec masking.
  int zero_tile =
      __builtin_amdgcn_readfirstlane((box_d2 >= LINE_WIDTH * LINE_WIDTH) ? 1 : 0);

  if (zero_tile) {
    // 16x16 tile of zeros: 2 x global_store_b128 per lane (64 float4 total).
    f4 z = {0.0f, 0.0f, 0.0f, 0.0f};
#pragma unroll
    for (int i = 0; i < 2; ++i) {
      int idx = lane * 2 + i;   // 0..63
      int r = idx >> 2;         // row within tile, 0..15
      int c4 = idx & 3;         // float4 column within row
      *(f4*)(out + (size_t)(y0 + r) * CANVAS_W + (size_t)(x0 + c4 * 4)) = z;
    }
    return;
  }

  // Rank-2 outer sum as a matrix product:
  //   dist2[m][n] = dy2[m]*1 + 1*dx2[n]  ->  D = A(16x4) x B(4x16) + 0
  // 32-bit A 16x4 layout: lanes 0-15 hold M=lane with VGPR0=K0, VGPR1=K1;
  // lanes 16-31 hold K2/K3 (zeros here). B is the symmetric 4x16 layout.
  v2f a, b;
  if (lane < 16) {
    float dy = (float)(y0 + lane) - py;
    float dx = (float)(x0 + lane) - px;
    a[0] = dy * dy;  // A[m][0] = dy^2
    a[1] = 1.0f;     // A[m][1] = 1
    b[0] = 1.0f;     // B[0][n] = 1
    b[1] = dx * dx;  // B[1][n] = dx^2
  } else {
    a[0] = 0.0f; a[1] = 0.0f;  // A[m][2..3] = 0
    b[0] = 0.0f; b[1] = 0.0f;  // B[2..3][n] = 0
  }

  v8f acc = {};
  acc = __builtin_amdgcn_wmma_f32_16x16x4_f32(
      /*neg_a=*/false, a, /*neg_b=*/false, b,
      /*c_mod=*/(short)0, acc, /*reuse_a=*/false, /*reuse_b=*/false);

  // max_distance = ||(4096,4096)|| computed exactly as the reference does.
  const float maxd =
      sqrtf((float)CANVAS_H * (float)CANVAS_H + (float)CANVAS_W * (float)CANVAS_W);

  // f32 C/D layout: VGPR v -> row M=v (lanes 0-15) or M=v+8 (lanes 16-31),
  // column N = lane & 15.
  const int n = lane & 15;
  const int mbase = (lane >> 4) * 8;
#pragma unroll
  for (int v = 0; v < 8; ++v) {
    float d = sqrtf(acc[v]);
    float val = 1.0f - (d / maxd + EPSILON_);
    float o = (d < LINE_WIDTH) ? val : 0.0f;
    out[(size_t)(y0 + mbase + v) * CANVAS_W + (size_t)(x0 + n)] = o;
  }
}

extern "C" void kernel_launch(void* const* d_in, const int* in_sizes, int n_in,
                              void* d_out, int out_size, void* d_ws, size_t ws_size,
                              hipStream_t stream) {
  const float* kp = (const float*)d_in[0];  // [1,2] normalized (y, x)
  float* out = (float*)d_out;               // [4096, 4096] f32

  // 8 waves/block, one 16x16 tile per wave: block covers 128x16 pixels.
  dim3 grid(CANVAS_W / 128, CANVAS_H / 16);
  PointGraphic2d_kernel<<<grid, 256, 0, stream>>>(kp, out);
}